// TripletCenterLoss_31911607009610
// MI455X (gfx1250) — compile-verified
//
#include <hip/hip_runtime.h>
#include <math.h>

// Problem constants (from reference)
#define BATCH 8192
#define FEAT  512
#define NCLS  40
#define NT    3            // 3 column tiles of 16 -> 48 padded classes
#define K0N   (FEAT / 4)   // 128 K-steps of 4 (V_WMMA_F32_16X16X4_F32)
#define MARGIN 1.0f
#define NBLK  (BATCH / 64) // 128 main-kernel blocks (4 waves x 16 rows each)

typedef float v2f __attribute__((ext_vector_type(2)));
typedef float v8f __attribute__((ext_vector_type(8)));

// Workspace layout (float index):
//   W[2..3]     class-presence bitmask (2 x u32)
//   W[4..43]    c2[c] = ||centers[c]||^2
//   W[128..255] per-block loss partials (deterministic reduction)
//   W[256..383] per-block prec partials
//   W[384..]    packed B fragments: 128 k0-steps x 3 tiles x 32 lanes x 2 floats
//               = 24576 floats (96 KB). Total ws need ~= 99.8 KB.
#define WS_PRES  2
#define WS_C2    4
#define WS_LOSSP 128
#define WS_PRECP 256
#define WS_BPACK 384

__global__ void tcl_init(float* W) {
  if (threadIdx.x < 2) ((unsigned*)(W + WS_PRES))[threadIdx.x] = 0u;
}

// Bitwise OR is order-independent -> deterministic despite atomics.
__global__ void tcl_presence(const int* __restrict__ tgt, float* W) {
  int i = blockIdx.x * blockDim.x + threadIdx.x;
  if (i < BATCH) {
    unsigned c = (unsigned)tgt[i] & 63u;
    atomicOr(((unsigned*)(W + WS_PRES)) + (c >> 5), 1u << (c & 31u));
  }
}

// One block per class: c2[c] = sum_k centers[c][k]^2 (fixed-order tree)
__global__ void tcl_c2(const float* __restrict__ centers, float* W) {
  __shared__ float s[256];
  const float* row = centers + blockIdx.x * FEAT;
  float a0 = row[threadIdx.x], a1 = row[threadIdx.x + 256];
  s[threadIdx.x] = a0 * a0 + a1 * a1;
  __syncthreads();
  for (int w = 128; w > 0; w >>= 1) {
    if (threadIdx.x < w) s[threadIdx.x] += s[threadIdx.x + w];
    __syncthreads();
  }
  if (threadIdx.x == 0) W[WS_C2 + blockIdx.x] = s[0];
}

// Pack centers^T into the 32-bit WMMA B-fragment lane layout:
// fragment for (k0, tile t): lane holds col c = t*16+(lane&15),
// VGPR v in {0,1} holds K = k0*4 + 2*(lane>>4) + v. Classes >= 40 are zero.
__global__ void tcl_packB(const float* __restrict__ centers, float* W) {
  int tid = blockIdx.x * blockDim.x + threadIdx.x;
  if (tid >= K0N * NT * 64) return;
  int v    = tid & 1;
  int lane = (tid >> 1) & 31;
  int kt   = tid >> 6;         // k0*3 + t
  int t    = kt % 3;
  int k0   = kt / 3;
  int c    = t * 16 + (lane & 15);
  int k    = k0 * 4 + 2 * (lane >> 4) + v;
  W[WS_BPACK + tid] = (c < NCLS) ? centers[c * FEAT + k] : 0.0f;
}

// Main: one wave per 16-row tile. D = inputs x centersPacked^T via fp32 WMMA,
// fused row-norm, hardest-positive/negative extraction, per-block partial sums.
__global__ void __launch_bounds__(128) tcl_main(const float* __restrict__ inp,
                                                const int* __restrict__ tgt,
                                                float* __restrict__ W) {
  const int lane = threadIdx.x & 31;
  const int wave = threadIdx.x >> 5;
  const int l15  = lane & 15;
  const int hi   = lane >> 4;
  const int Mbase = (blockIdx.x * 4 + wave) * 16;

  const float* bpack = W + WS_BPACK;
  v8f acc0 = {}, acc1 = {}, acc2 = {};
  float xp = 0.f;

  // A-fragment addressing (32-bit A 16x4): lane -> row l15,
  // VGPR0/1 -> K = k0*4 + 2*hi + {0,1}
  const float* arow = inp + (Mbase + l15) * FEAT + 2 * hi;
  const float* brow = bpack + lane * 2;

#pragma unroll 4
  for (int k0 = 0; k0 < K0N; ++k0) {
    v2f a  = *(const v2f*)(arow + k0 * 4);
    v2f b0 = *(const v2f*)(brow + (k0 * 3 + 0) * 64);
    v2f b1 = *(const v2f*)(brow + (k0 * 3 + 1) * 64);
    v2f b2 = *(const v2f*)(brow + (k0 * 3 + 2) * 64);
    __builtin_prefetch(arow + k0 * 4 + 64, 0, 3);   // WGP-scope global_prefetch_b8
    xp += a.x * a.x + a.y * a.y;
    acc0 = __builtin_amdgcn_wmma_f32_16x16x4_f32(false, a, false, b0, (short)0, acc0, false, false);
    acc1 = __builtin_amdgcn_wmma_f32_16x16x4_f32(false, a, false, b1, (short)0, acc1, false, false);
    acc2 = __builtin_amdgcn_wmma_f32_16x16x4_f32(false, a, false, b2, (short)0, acc2, false, false);
  }

  // x2: lane l covered K mod4 in {0,1}(lo half) or {2,3}(hi half) for row l15.
  xp += __shfl_xor(xp, 16, 32);          // now every lane l holds x2[row l15]
  int tl = tgt[Mbase + l15];             // lane l holds target[row l15]

  const unsigned* pres = (const unsigned*)(W + WS_PRES);
  unsigned p0 = pres[0], p1 = pres[1];

  float c2v[NT];
  bool  vld[NT];
  int   cls[NT];
#pragma unroll
  for (int t = 0; t < NT; ++t) {
    int c = t * 16 + l15;
    cls[t] = c;
    unsigned bit = (c < 32) ? ((p0 >> c) & 1u) : ((p1 >> (c - 32)) & 1u);
    vld[t] = (c < NCLS) && (bit != 0u);
    c2v[t] = (c < NCLS) ? W[WS_C2 + c] : 0.f;
  }

  const float INF = __builtin_inff();
  float lossAcc = 0.f, precAcc = 0.f;

#pragma unroll
  for (int e = 0; e < 8; ++e) {
    // C/D layout: VGPR e, lanes 0-15 -> row e; lanes 16-31 -> row e+8; col = l15
    int   m   = e + 8 * hi;
    float x2m = __shfl(xp, m, 32);
    int   tm  = __shfl(tl, m, 32);

    float dots[NT] = { acc0[e], acc1[e], acc2[e] };
    float ap = -INF, an = INF;
#pragma unroll
    for (int t = 0; t < NT; ++t) {
      float d2 = x2m + c2v[t] - 2.0f * dots[t];
      float d  = sqrtf(fmaxf(d2, 1e-12f));
      if (vld[t]) {
        if (cls[t] == tm) ap = d;           // hardest positive == own-class dist
        else              an = fminf(an, d);
      }
    }
    // reduce across the 16 lanes of this half (xor masks < 16 stay in-half)
#pragma unroll
    for (int w = 1; w < 16; w <<= 1) {
      ap = fmaxf(ap, __shfl_xor(ap, w, 32));
      an = fminf(an, __shfl_xor(an, w, 32));
    }
    if (l15 == 0) {
      lossAcc += fmaxf(ap - an + MARGIN, 0.f);
      precAcc += (an > ap) ? 1.f : 0.f;
    }
  }

  // Deterministic block partial: lane0 + lane16 per wave, then LDS across waves.
  lossAcc += __shfl_xor(lossAcc, 16, 32);
  precAcc += __shfl_xor(precAcc, 16, 32);
  __shared__ float sl[4], sp[4];
  if (lane == 0) { sl[wave] = lossAcc; sp[wave] = precAcc; }
  __syncthreads();
  if (threadIdx.x == 0) {
    W[WS_LOSSP + blockIdx.x] = (sl[0] + sl[1]) + (sl[2] + sl[3]);
    W[WS_PRECP + blockIdx.x] = (sp[0] + sp[1]) + (sp[2] + sp[3]);
  }
}

// Fixed-order tree reduction of 128 block partials -> deterministic output.
__global__ void __launch_bounds__(128) tcl_fin(const float* __restrict__ W,
                                               float* __restrict__ out) {
  __shared__ float sl[128], sp[128];
  sl[threadIdx.x] = W[WS_LOSSP + threadIdx.x];
  sp[threadIdx.x] = W[WS_PRECP + threadIdx.x];
  __syncthreads();
  for (int w = 64; w > 0; w >>= 1) {
    if (threadIdx.x < w) {
      sl[threadIdx.x] += sl[threadIdx.x + w];
      sp[threadIdx.x] += sp[threadIdx.x + w];
    }
    __syncthreads();
  }
  if (threadIdx.x == 0) {
    out[0] = sl[0] / (float)BATCH;
    out[1] = sp[0] / (float)BATCH;
  }
}

extern "C" void kernel_launch(void* const* d_in, const int* in_sizes, int n_in,
                              void* d_out, int out_size, void* d_ws, size_t ws_size,
                              hipStream_t stream) {
  const float* inputs  = (const float*)d_in[0];
  const int*   targets = (const int*)d_in[1];
  const float* centers = (const float*)d_in[2];
  float* W   = (float*)d_ws;
  float* out = (float*)d_out;

  tcl_init<<<1, 32, 0, stream>>>(W);
  tcl_presence<<<(BATCH + 255) / 256, 256, 0, stream>>>(targets, W);
  tcl_c2<<<NCLS, 256, 0, stream>>>(centers, W);
  tcl_packB<<<(K0N * NT * 64 + 255) / 256, 256, 0, stream>>>(centers, W);
  tcl_main<<<NBLK, 128, 0, stream>>>(inputs, targets, W);
  tcl_fin<<<1, 128, 0, stream>>>(W, out);
}